// CrossEntropy_42021960024720
// MI455X (gfx1250) — compile-verified
//
#include <hip/hip_runtime.h>
#include <math.h>

// Cross-entropy: loss = -mean_n( logit[n,label] - max_n - log(sum_c exp(logit-max)) )
// N = 262144 rows, C = 1000 classes, fp32. Pure HBM-bandwidth problem:
// ~1.05 GB read once -> ~45 us floor at 23.3 TB/s. Strategy: wave-per-row
// streaming through the gfx1250 async global->LDS DMA path, double-buffered.

#define CE_C      1000
#define CE_C4     250        // float4 chunks per row
#define CE_SLOTS  8          // ceil(250 / 32 lanes)
#define CE_BLOCK  256        // 8 waves (wave32)
#define CE_WAVES  8
#define CE_GRID   8192       // 65536 waves total, 4 rows per wave (grid-stride)
#define CE_CPOL_NT 1         // TH[2:0] = 1 = NT: stream, don't pollute 192MB L2

typedef float v4f __attribute__((ext_vector_type(4)));
typedef int   v4i __attribute__((__vector_size__(4 * sizeof(int))));  // matches builtin param

// Probe for the gfx1250 async global->LDS path.
#if defined(__has_builtin)
#  if __has_builtin(__builtin_amdgcn_global_load_async_to_lds_b128) && \
      __has_builtin(__builtin_amdgcn_s_wait_asynccnt)
#    define CE_HAVE_ASYNC 1
#  endif
#endif
#ifndef CE_HAVE_ASYNC
#  define CE_HAVE_ASYNC 0
#endif

#if CE_HAVE_ASYNC
typedef __attribute__((address_space(1))) v4i* gptr_v4i;
typedef __attribute__((address_space(3))) v4i* lptr_v4i;

// Issue the 8 async B128 loads that stage one 1000-float row into LDS.
// Always issues exactly 8 instructions per wave (k=7 runs with partial EXEC),
// so ASYNCcnt advances by 8 per staged row.
__device__ __forceinline__ void ce_stage_row(const float* __restrict__ rowp,
                                             float* dst, int lane)
{
    #pragma unroll
    for (int k = 0; k < CE_SLOTS; ++k) {
        int j = lane + 32 * k;
        if (j < CE_C4) {
            __builtin_amdgcn_global_load_async_to_lds_b128(
                (gptr_v4i)(rowp + 4 * j),
                (lptr_v4i)(dst + 4 * j),
                0, CE_CPOL_NT);
        }
    }
}
#endif

__global__ __launch_bounds__(CE_BLOCK) void ce_rows_kernel(
    const float* __restrict__ logits,
    const int*   __restrict__ labels,
    double*      __restrict__ acc,
    int n_rows)
{
    const int lane  = threadIdx.x & 31;
    const int wave  = threadIdx.x >> 5;
    const int gwave = blockIdx.x * CE_WAVES + wave;
    const int nwave = gridDim.x * CE_WAVES;

    __shared__ double bsum[CE_WAVES];
#if CE_HAVE_ASYNC
    __shared__ float stage[CE_WAVES][2][1024];   // 2 x 4KB per wave, 64KB/block
#endif

    const v4f vninf = { -INFINITY, -INFINITY, -INFINITY, -INFINITY };
    double wacc = 0.0;

#if CE_HAVE_ASYNC
    // ---------------- double-buffered async pipeline ----------------
    int cur = 0;
    if (gwave < n_rows)
        ce_stage_row(logits + (size_t)gwave * CE_C, &stage[wave][0][0], lane);

    for (int r = gwave; r < n_rows; r += nwave) {
        const int rn = r + nwave;
        if (rn < n_rows) {
            // Prefetch next row into the other buffer, then wait until only
            // those 8 requests remain outstanding -> current row has landed.
            ce_stage_row(logits + (size_t)rn * CE_C, &stage[wave][cur ^ 1][0], lane);
            __builtin_amdgcn_s_wait_asynccnt(CE_SLOTS);
        } else {
            __builtin_amdgcn_s_wait_asynccnt(0);
        }

        float* mystage = &stage[wave][cur][0];
        v4f v[CE_SLOTS];
        #pragma unroll
        for (int k = 0; k < CE_SLOTS; ++k) {
            int j = lane + 32 * k;
            v[k] = (j < CE_C4) ? *(const v4f*)(mystage + 4 * j) : vninf;
        }
#else
    // ---------------- fallback: direct non-temporal loads ----------------
    for (int r = gwave; r < n_rows; r += nwave) {
        const float* rowp = logits + (size_t)r * CE_C;
        v4f v[CE_SLOTS];
        #pragma unroll
        for (int k = 0; k < CE_SLOTS; ++k) {
            int j = lane + 32 * k;
            v[k] = (j < CE_C4)
                     ? __builtin_nontemporal_load((const v4f*)(rowp + 4 * j))
                     : vninf;
        }
#endif

        // ---- row max (registers, then 5-step wave32 butterfly) ----
        float m = -INFINITY;
        #pragma unroll
        for (int k = 0; k < CE_SLOTS; ++k) {
            m = fmaxf(m, fmaxf(fmaxf(v[k].x, v[k].y), fmaxf(v[k].z, v[k].w)));
        }
        #pragma unroll
        for (int off = 16; off > 0; off >>= 1)
            m = fmaxf(m, __shfl_xor(m, off, 32));

        // ---- sum of exp(x - m): -inf padding contributes exp(-inf)=0 ----
        float s = 0.0f;
        #pragma unroll
        for (int k = 0; k < CE_SLOTS; ++k) {
            s += __expf(v[k].x - m);
            s += __expf(v[k].y - m);
            s += __expf(v[k].z - m);
            s += __expf(v[k].w - m);
        }
        #pragma unroll
        for (int off = 16; off > 0; off >>= 1)
            s += __shfl_xor(s, off, 32);

        // ---- contribution: -(x_label - m - log(s)) = m + log(s) - x_label ----
        if (lane == 0) {
            int lbl = labels[r];
#if CE_HAVE_ASYNC
            float xl = mystage[lbl];
#else
            float xl = rowp[lbl];
#endif
            wacc += (double)m + (double)__logf(s) - (double)xl;
        }
#if CE_HAVE_ASYNC
        cur ^= 1;
#endif
    }

    // ---- block reduction: 8 wave partials -> one f64 atomic per block ----
    if (lane == 0) bsum[wave] = wacc;
    __syncthreads();
    if (threadIdx.x == 0) {
        double t = 0.0;
        #pragma unroll
        for (int i = 0; i < CE_WAVES; ++i) t += bsum[i];
        atomicAdd(acc, t);   // global_atomic_add_f64
    }
}

__global__ void ce_finalize_kernel(const double* __restrict__ acc,
                                   float* __restrict__ out, int n_rows)
{
    if (threadIdx.x == 0 && blockIdx.x == 0)
        out[0] = (float)(acc[0] / (double)n_rows);
}

extern "C" void kernel_launch(void* const* d_in, const int* in_sizes, int n_in,
                              void* d_out, int out_size, void* d_ws, size_t ws_size,
                              hipStream_t stream)
{
    const float* logits = (const float*)d_in[0];
    const int*   labels = (const int*)d_in[1];
    double*      acc    = (double*)d_ws;
    float*       out    = (float*)d_out;
    const int n_rows    = in_sizes[1];     // labels: one per row

    (void)hipMemsetAsync(acc, 0, sizeof(double), stream);
    ce_rows_kernel<<<CE_GRID, CE_BLOCK, 0, stream>>>(logits, labels, acc, n_rows);
    ce_finalize_kernel<<<1, 32, 0, stream>>>(acc, out, n_rows);
}